// TopologicalSignature_79671643341164
// MI455X (gfx1250) — compile-verified
//
#include <hip/hip_runtime.h>
#include <math.h>

#define NPTS 512
#define DIM  512

typedef __attribute__((ext_vector_type(2))) float v2f;
typedef __attribute__((ext_vector_type(8))) float v8f;

// ---------------------------------------------------------------------------
// Kernel 1: squared row norms  nrm[i] = sum_j x[i][j]^2
// ---------------------------------------------------------------------------
__global__ __launch_bounds__(128) void row_norms_kernel(const float* __restrict__ x,
                                                        float* __restrict__ nrm) {
  __shared__ float red[128];
  const int row = blockIdx.x;
  float s = 0.0f;
  for (int j = threadIdx.x; j < DIM; j += 128) {
    float v = x[row * DIM + j];
    s += v * v;
  }
  red[threadIdx.x] = s;
  __syncthreads();
  for (int off = 64; off > 0; off >>= 1) {
    if (threadIdx.x < off) red[threadIdx.x] += red[threadIdx.x + off];
    __syncthreads();
  }
  if (threadIdx.x == 0) nrm[row] = red[0];
}

// ---------------------------------------------------------------------------
// Kernel 2: distance matrix via fp32 WMMA Gram tiles.
// One wave (32 threads) per 16x16 output tile. D = A(16x4) x B(4x16) + C,
// accumulated over K = 512 in steps of 4 (128 v_wmma_f32_16x16x4_f32 ops).
//
// fp32 A layout (ISA 7.12.2): lanes 0-15: VGPR0=K0, VGPR1=K1 for M=lane;
// lanes 16-31: VGPR0=K2, VGPR1=K3 for M=lane-16.  So lane loads the float2
// at X[m0+lr][k + 2*half].  B (= X^T columns of the same X) mirrors this
// with N in place of M, giving an identical gather with n0.
// ---------------------------------------------------------------------------
__global__ __launch_bounds__(32) void dist_wmma_kernel(const float* __restrict__ x,
                                                       const float* __restrict__ nrm,
                                                       float* __restrict__ dist) {
  const int m0   = blockIdx.x * 16;
  const int n0   = blockIdx.y * 16;
  const int lane = threadIdx.x;
  const int half = lane >> 4;   // 0 or 1
  const int lr   = lane & 15;

  const float* arow = x + (size_t)(m0 + lr) * DIM + 2 * half;
  const float* brow = x + (size_t)(n0 + lr) * DIM + 2 * half;

  v8f acc = {0.f, 0.f, 0.f, 0.f, 0.f, 0.f, 0.f, 0.f};

#pragma unroll 4
  for (int k = 0; k < DIM; k += 4) {
    v2f a = *(const v2f*)(arow + k);
    v2f b = *(const v2f*)(brow + k);
    // 8 args: (neg_a, A, neg_b, B, c_mod, C, reuse_a, reuse_b)
    acc = __builtin_amdgcn_wmma_f32_16x16x4_f32(false, a, false, b,
                                                (short)0, acc, false, false);
  }

  // C/D layout: VGPR r holds M = m0 + r + 8*half, N = n0 + lr.
  const int   n  = n0 + lr;
  const float nn = nrm[n];
#pragma unroll
  for (int r = 0; r < 8; ++r) {
    const int   m  = m0 + r + 8 * half;
    const float d2 = nrm[m] + nn - 2.0f * acc[r];
    dist[(size_t)m * NPTS + n] = sqrtf(fmaxf(d2, 0.0f));
  }
}

// ---------------------------------------------------------------------------
// Kernel 3: Prim's MST (O(n^2)) + bitonic sort of the 511 edge weights.
// Sorted-ascending MST edge weights == Kruskal acceptance-order distances
// == reference output.
// ---------------------------------------------------------------------------
__global__ __launch_bounds__(NPTS) void prim_sort_kernel(const float* __restrict__ dist,
                                                         float* __restrict__ out) {
  __shared__ float minD[NPTS];
  __shared__ float sval[NPTS];
  __shared__ int   sidx[NPTS];
  __shared__ int   inTree[NPTS];
  __shared__ float edgeW[NPTS];   // 511 edges + 1 pad slot
  __shared__ int   bestShared;

  const int tid = threadIdx.x;
  inTree[tid] = (tid == 0) ? 1 : 0;
  minD[tid]   = dist[tid];        // row 0 = distances from vertex 0
  __syncthreads();

  for (int it = 0; it < NPTS - 1; ++it) {
    sval[tid] = inTree[tid] ? __builtin_inff() : minD[tid];
    sidx[tid] = tid;
    __syncthreads();
    for (int s = NPTS / 2; s > 0; s >>= 1) {
      if (tid < s) {
        if (sval[tid + s] < sval[tid]) {
          sval[tid] = sval[tid + s];
          sidx[tid] = sidx[tid + s];
        }
      }
      __syncthreads();
    }
    if (tid == 0) {
      const int b = sidx[0];
      edgeW[it]   = sval[0];
      inTree[b]   = 1;
      bestShared  = b;
    }
    __syncthreads();
    const int b = bestShared;
    if (!inTree[tid]) {
      const float dv = dist[(size_t)b * NPTS + tid];
      if (dv < minD[tid]) minD[tid] = dv;
    }
    __syncthreads();
  }

  if (tid == 0) edgeW[NPTS - 1] = __builtin_inff();   // pad to power of two
  __syncthreads();

  // Bitonic sort of 512 values, ascending.
  for (int k = 2; k <= NPTS; k <<= 1) {
    for (int j = k >> 1; j > 0; j >>= 1) {
      const int ixj = tid ^ j;
      if (ixj > tid) {
        const bool  up = ((tid & k) == 0);
        const float a  = edgeW[tid];
        const float c  = edgeW[ixj];
        if ((a > c) == up) {
          edgeW[tid] = c;
          edgeW[ixj] = a;
        }
      }
      __syncthreads();
    }
  }

  if (tid < NPTS - 1) out[tid] = edgeW[tid];
}

// ---------------------------------------------------------------------------
extern "C" void kernel_launch(void* const* d_in, const int* in_sizes, int n_in,
                              void* d_out, int out_size, void* d_ws, size_t ws_size,
                              hipStream_t stream) {
  (void)in_sizes; (void)n_in; (void)out_size; (void)ws_size;

  const float* x   = (const float*)d_in[0];
  float*       ws  = (float*)d_ws;
  float*       dm  = ws;                          // NPTS*NPTS floats (1 MB)
  float*       nrm = ws + (size_t)NPTS * NPTS;    // NPTS floats

  row_norms_kernel<<<dim3(NPTS), dim3(128), 0, stream>>>(x, nrm);
  dist_wmma_kernel<<<dim3(NPTS / 16, NPTS / 16), dim3(32), 0, stream>>>(x, nrm, dm);
  prim_sort_kernel<<<dim3(1), dim3(NPTS), 0, stream>>>(dm, (float*)d_out);
}